// Classifier_4535485464940
// MI455X (gfx1250) — compile-verified
//
#include <hip/hip_runtime.h>
#include <hip/hip_bf16.h>
#include <math.h>

// Problem constants from reference
#define Bn   64
#define Ln   2048
#define Dn   300
#define Cn   2
#define Kn   111
#define PADn 55

typedef __attribute__((ext_vector_type(2))) float v2f;
typedef __attribute__((ext_vector_type(8))) float v8f;

// ---------------------------------------------------------------------------
// Phase 1: g[b,c,l] = dot(cls_n[c], emb[token]) / max(||emb||, 1e-12)
// One wave per 16-token tile; 75 chained V_WMMA_F32_16X16X4_F32 per tile.
// ---------------------------------------------------------------------------
__global__ __launch_bounds__(256)
void k_phase1(const int* __restrict__ tok, const float* __restrict__ emb,
              const float* __restrict__ cls, float* __restrict__ g)
{
    // Normalized class vectors in LDS: rows 0,1 = classes, row 2 = zeros
    // (lanes with N>=2 read the zero row so EXEC stays full, branch-free).
    __shared__ float clsn[3 * 304];
    __shared__ float red[256];
    const int tid = threadIdx.x;

    for (int i = tid; i < 3 * 304; i += 256) clsn[i] = 0.0f;
    __syncthreads();
    for (int c = 0; c < Cn; ++c) {
        float s = 0.0f;
        for (int d = tid; d < Dn; d += 256) { float v = cls[c * Dn + d]; s += v * v; }
        red[tid] = s;
        __syncthreads();
        for (int off = 128; off > 0; off >>= 1) {
            if (tid < off) red[tid] += red[tid + off];
            __syncthreads();
        }
        float rn = 1.0f / fmaxf(sqrtf(red[0]), 1e-12f);
        __syncthreads();
        for (int d = tid; d < Dn; d += 256) clsn[c * 304 + d] = cls[c * Dn + d] * rn;
        __syncthreads();
    }

    const int wave = tid >> 5;
    const int lane = tid & 31;
    const int tileId = blockIdx.x * 8 + wave;   // 8192 tiles total
    const int b  = tileId >> 7;                  // / (L/16)
    const int l0 = (tileId & 127) * 16;
    const int m    = lane & 15;                  // token row this lane feeds
    const int hi   = lane >> 4;                  // 0: K%4 in {0,1}, 1: {2,3}
    const int koff = hi * 2;

    const int token = tok[b * Ln + l0 + m];
    const float* __restrict__ row = emb + (long)token * Dn;
    const float* __restrict__ bb  = &clsn[(m < 2 ? m : 2) * 304];

    v8f acc = {};
    float n2 = 0.0f;
    #pragma unroll 5
    for (int k = 0; k < 75; ++k) {
        const int d0 = 4 * k + koff;
        v2f a;  a.x = row[d0];           a.y = row[d0 + 1];
        v2f bf; bf.x = bb[d0];           bf.y = bb[d0 + 1];
        n2 += a.x * a.x + a.y * a.y;
        // V_WMMA_F32_16X16X4_F32: D = A(16x4) * B(4x16) + C
        acc = __builtin_amdgcn_wmma_f32_16x16x4_f32(
            false, a, false, bf, (short)0, acc, false, false);
    }

    // Full per-token norm^2 = this lane's half + partner lane's half.
    n2 += __shfl_xor(n2, 16, 32);
    // Lane j (and j+16) now holds norm^2 of token j (j = lane&15).

    const int  c      = m;                 // class index (valid when m<2)
    const bool active = (m < 2);
    const int  mbase  = hi * 8;            // C-frag element i -> token mbase+i
    float* __restrict__ gout = g + (((long)b * Cn + c) * Ln + l0 + mbase);
    #pragma unroll
    for (int i = 0; i < 8; ++i) {
        float nn    = __shfl(n2, mbase + i, 32);
        float scale = 1.0f / fmaxf(sqrtf(nn), 1e-12f);
        if (active) gout[i] = acc[i] * scale;
    }
}

// ---------------------------------------------------------------------------
// Phase 2: conv1d(C->C,k=111,pad=55) + bias + relu, max over channels,
// softmax over L. One block per batch row, everything staged in LDS.
// ---------------------------------------------------------------------------
__global__ __launch_bounds__(256)
void k_phase2(const float* __restrict__ g, const float* __restrict__ conv_w,
              const float* __restrict__ conv_b, float* __restrict__ betaOut)
{
    __shared__ float gin[Cn][Ln + 2 * PADn + 2];   // zero-padded input, 2160/row
    __shared__ float wsm[Cn * Cn * Kn];            // 444 weights
    __shared__ float red[256];
    const int b = blockIdx.x, tid = threadIdx.x;

    for (int i = tid; i < Cn * (Ln + 2 * PADn + 2); i += 256) (&gin[0][0])[i] = 0.0f;
    __syncthreads();
    for (int i = tid; i < Ln; i += 256) {
        gin[0][PADn + i] = g[((long)b * Cn + 0) * Ln + i];
        gin[1][PADn + i] = g[((long)b * Cn + 1) * Ln + i];
    }
    for (int i = tid; i < Cn * Cn * Kn; i += 256) wsm[i] = conv_w[i];
    __syncthreads();

    const float cb0 = conv_b[0], cb1 = conv_b[1];
    float h[8];
    #pragma unroll
    for (int j = 0; j < 8; ++j) {
        const int x = tid + 256 * j;
        float a0 = 0.0f, a1 = 0.0f;
        for (int k = 0; k < Kn; ++k) {
            const float i0 = gin[0][x + k];
            const float i1 = gin[1][x + k];
            a0 += wsm[0 * Kn + k] * i0 + wsm[1 * Kn + k] * i1;   // out ch 0
            a1 += wsm[2 * Kn + k] * i0 + wsm[3 * Kn + k] * i1;   // out ch 1
        }
        h[j] = fmaxf(fmaxf(a0 + cb0, 0.0f), fmaxf(a1 + cb1, 0.0f));
    }

    // softmax over the 2048 positions
    float lm = -INFINITY;
    #pragma unroll
    for (int j = 0; j < 8; ++j) lm = fmaxf(lm, h[j]);
    red[tid] = lm; __syncthreads();
    for (int off = 128; off > 0; off >>= 1) {
        if (tid < off) red[tid] = fmaxf(red[tid], red[tid + off]);
        __syncthreads();
    }
    const float M = red[0];
    __syncthreads();

    float ls = 0.0f;
    #pragma unroll
    for (int j = 0; j < 8; ++j) { h[j] = __expf(h[j] - M); ls += h[j]; }
    red[tid] = ls; __syncthreads();
    for (int off = 128; off > 0; off >>= 1) {
        if (tid < off) red[tid] += red[tid + off];
        __syncthreads();
    }
    const float inv = 1.0f / red[0];
    #pragma unroll
    for (int j = 0; j < 8; ++j)
        betaOut[(long)b * Ln + tid + 256 * j] = h[j] * inv;
}

// ---------------------------------------------------------------------------
// Phase 3: z[b,:] = sum_l beta[b,l]*emb[token], logits = log_softmax(z@W^T+b)
// One block per batch row; thread d accumulates dimension d. Re-gathers the
// (L2-resident) table instead of spilling 157MB of emb.
// ---------------------------------------------------------------------------
__global__ __launch_bounds__(320)
void k_phase3(const int* __restrict__ tok, const float* __restrict__ emb,
              const float* __restrict__ beta, const float* __restrict__ lin_w,
              const float* __restrict__ lin_b, float* __restrict__ logits)
{
    __shared__ float s_acc[2];
    const int b = blockIdx.x, tid = threadIdx.x;
    if (tid < 2) s_acc[tid] = 0.0f;

    const int*   __restrict__ trow = tok  + (long)b * Ln;
    const float* __restrict__ brow = beta + (long)b * Ln;
    const bool act = (tid < Dn);
    const int  dd  = act ? tid : 0;

    float z = 0.0f;
    for (int l = 0; l < Ln; l += 4) {
        const int t0 = trow[l],     t1 = trow[l + 1];
        const int t2 = trow[l + 2], t3 = trow[l + 3];
        const float b0 = brow[l],     b1 = brow[l + 1];
        const float b2 = brow[l + 2], b3 = brow[l + 3];
        z += b0 * emb[(long)t0 * Dn + dd];
        z += b1 * emb[(long)t1 * Dn + dd];
        z += b2 * emb[(long)t2 * Dn + dd];
        z += b3 * emb[(long)t3 * Dn + dd];
    }
    __syncthreads();
    if (act) {
        atomicAdd(&s_acc[0], z * lin_w[0 * Dn + tid]);
        atomicAdd(&s_acc[1], z * lin_w[1 * Dn + tid]);
    }
    __syncthreads();
    if (tid == 0) {
        const float o0 = s_acc[0] + lin_b[0];
        const float o1 = s_acc[1] + lin_b[1];
        const float mm = fmaxf(o0, o1);
        const float lse = mm + __logf(__expf(o0 - mm) + __expf(o1 - mm));
        logits[b * 2 + 0] = o0 - lse;
        logits[b * 2 + 1] = o1 - lse;
    }
}

// ---------------------------------------------------------------------------
extern "C" void kernel_launch(void* const* d_in, const int* in_sizes, int n_in,
                              void* d_out, int out_size, void* d_ws, size_t ws_size,
                              hipStream_t stream)
{
    const int*   tok = (const int*)  d_in[0];
    const float* emb = (const float*)d_in[1];
    const float* cls = (const float*)d_in[2];
    const float* cw  = (const float*)d_in[3];
    const float* cb  = (const float*)d_in[4];
    const float* lw  = (const float*)d_in[5];
    const float* lb  = (const float*)d_in[6];

    float* out    = (float*)d_out;
    float* logits = out;               // [B, C] = 128 floats
    float* beta   = out + Bn * Cn;     // [B, L, 1] = 131072 floats
    float* g      = (float*)d_ws;      // [B, C, L] = 1 MB scratch

    // 8192 tiles of 16 tokens, 8 waves/block -> 1024 blocks
    k_phase1<<<1024, 256, 0, stream>>>(tok, emb, cls, g);
    k_phase2<<<Bn, 256, 0, stream>>>(g, cw, cb, beta);
    k_phase3<<<Bn, 320, 0, stream>>>(tok, emb, beta, lw, lb, logits);
}